// GuidedFilter_3358664425939
// MI455X (gfx1250) — compile-verified
//
#include <hip/hip_runtime.h>

// Guided filter (R=40, EPS=1e-3) on gfx1250.
// Box filter expressed as banded GEMM; horizontal pass runs on
// V_WMMA_F32_16X16X4_F32, vertical pass is a VALU sliding window.
// Two kernels: pass1 (I,p,Ip,II -> a,b in d_ws), pass2 (a,b -> out, fused with I).

#define Hh 1024
#define Ww 1024
#define RAD 40
#define STRIP 64                 // output columns per block
#define HALO (STRIP + 2 * RAD)   // 144 halo columns
#define SEGH 128                 // output rows per block
#define VSTRIDE 148              // padded LDS row stride (floats) -> conflict-free A reads

typedef __attribute__((ext_vector_type(2))) float v2f;
typedef __attribute__((ext_vector_type(8))) float v8f;

__device__ __forceinline__ int imin(int a, int b) { return a < b ? a : b; }
__device__ __forceinline__ int imax(int a, int b) { return a > b ? a : b; }

template <int PASS>
__global__ __launch_bounds__(32)
void guided_boxpass(const float* __restrict__ s0,   // I (pass1) / a (pass2)
                    const float* __restrict__ s1,   // p (pass1) / b (pass2)
                    float* __restrict__ d0,         // a (pass1) / out (pass2)
                    float* __restrict__ d1,         // b (pass1) / unused
                    const float* __restrict__ guide) // unused / I
{
    constexpr int NQ = (PASS == 1) ? 4 : 2;
    __shared__ __align__(16) float V[NQ][16][VSTRIDE];

    const int lane  = threadIdx.x;            // wave32: 0..31, EXEC all-ones
    const int x0    = blockIdx.x * STRIP;
    const int y0    = blockIdx.y * SEGH;
    const int plane = blockIdx.z;
    const size_t off = (size_t)plane * Hh * Ww;
    const float* q0 = s0 + off;
    const float* q1 = s1 + off;

    // ---------------- warm-up: vertical window sum for row (y0-1) -> slot 15 ----
    for (int c = lane; c < HALO; c += 32) {
        float acc[NQ];
#pragma unroll
        for (int q = 0; q < NQ; ++q) acc[q] = 0.f;
        const int xa = x0 - RAD + c;
        if (xa >= 0 && xa < Ww) {
            for (int r = y0 - RAD - 1; r <= y0 + RAD - 1; ++r) {
                if (r >= 0 && r < Hh) {
                    const float a0 = q0[(size_t)r * Ww + xa];
                    const float a1 = q1[(size_t)r * Ww + xa];
                    acc[0] += a0; acc[1] += a1;
                    if constexpr (PASS == 1) { acc[2] += a0 * a1; acc[3] += a0 * a0; }
                }
            }
        }
#pragma unroll
        for (int q = 0; q < NQ; ++q) V[q][15][c] = acc[q];
    }

    const int h  = lane >> 4;   // K-half for A/B operands, M-half for C/D
    const int nl = lane & 15;   // M for A, N for B/C/D

    for (int Y = y0; Y < y0 + SEGH; Y += 16) {
        // ---------- stage A: vertical sliding-window sums for rows Y..Y+15 ----
        for (int m = 0; m < 16; ++m) {
            const int y    = Y + m;
            const int prev = (m == 0) ? 15 : (m - 1);
            const int rAdd = y + RAD;       // always >= 0
            const int rSub = y - RAD - 1;   // always < Hh
            for (int c = lane; c < HALO; c += 32) {
                const int xa = x0 - RAD + c;
                float addv[NQ], subv[NQ];
#pragma unroll
                for (int q = 0; q < NQ; ++q) { addv[q] = 0.f; subv[q] = 0.f; }
                if (xa >= 0 && xa < Ww) {
                    if (rAdd < Hh) {
                        const float a0 = q0[(size_t)rAdd * Ww + xa];
                        const float a1 = q1[(size_t)rAdd * Ww + xa];
                        addv[0] = a0; addv[1] = a1;
                        if constexpr (PASS == 1) { addv[2] = a0 * a1; addv[3] = a0 * a0; }
                        if (c < 32 && rAdd + 16 < Hh)   // stream next rows (L2-resident)
                            __builtin_prefetch(&q0[(size_t)(rAdd + 16) * Ww + xa], 0, 1);
                    }
                    if (rSub >= 0) {
                        const float a0 = q0[(size_t)rSub * Ww + xa];
                        const float a1 = q1[(size_t)rSub * Ww + xa];
                        subv[0] = a0; subv[1] = a1;
                        if constexpr (PASS == 1) { subv[2] = a0 * a1; subv[3] = a0 * a0; }
                    }
                }
#pragma unroll
                for (int q = 0; q < NQ; ++q)
                    V[q][m][c] = V[q][prev][c] + addv[q] - subv[q];
            }
        }

        // ---------- stage B: horizontal box sum = banded GEMM on WMMA ---------
        for (int t = 0; t < 4; ++t) {
            v8f acc[NQ];
#pragma unroll
            for (int q = 0; q < NQ; ++q) {
#pragma unroll
                for (int e = 0; e < 8; ++e) acc[q][e] = 0.f;
            }
            for (int s = 0; s < 24; ++s) {
                const int k0 = 4 * s + 2 * h;         // K of lane's first element
                v2f Bv;                                // band: B[k][n] = (n <= k <= n+80)
                Bv.x = (k0     >= nl && k0     <= nl + 2 * RAD) ? 1.f : 0.f;
                Bv.y = (k0 + 1 >= nl && k0 + 1 <= nl + 2 * RAD) ? 1.f : 0.f;
#pragma unroll
                for (int q = 0; q < NQ; ++q) {
                    // A = data: lane holds row m=nl, K slice cols (8B-aligned LDS read)
                    const v2f Av = *(const v2f*)&V[q][nl][16 * t + k0];
                    acc[q] = __builtin_amdgcn_wmma_f32_16x16x4_f32(
                                 false, Av, false, Bv, (short)0, acc[q], false, false);
                }
            }
            // ------- per-pixel combine + store (C/D layout: vgpr j -> M=j|j+8) --
            const int xx = x0 + 16 * t + nl;
            const float cx = (float)(imin(xx + RAD, Ww - 1) - imax(xx - RAD, 0) + 1);
#pragma unroll
            for (int j = 0; j < 8; ++j) {
                const int yy = Y + j + 8 * h;
                const float cy  = (float)(imin(yy + RAD, Hh - 1) - imax(yy - RAD, 0) + 1);
                const float inv = 1.f / (cx * cy);
                const size_t idx = off + (size_t)yy * Ww + xx;
                if constexpr (PASS == 1) {
                    const float mI  = acc[0][j] * inv;
                    const float mp  = acc[1][j] * inv;
                    const float mIp = acc[2][j] * inv;
                    const float mII = acc[3][j] * inv;
                    const float varI = mII - mI * mI;
                    const float aP   = (mIp - mI * mp) / (varI + 1e-3f);
                    d0[idx] = aP;
                    d1[idx] = mp - aP * mI;
                } else {
                    const float ma = acc[0][j] * inv;
                    const float mb = acc[1][j] * inv;
                    d0[idx] = ma * guide[idx] + mb;
                }
            }
        }
    }
}

extern "C" void kernel_launch(void* const* d_in, const int* in_sizes, int n_in,
                              void* d_out, int out_size, void* d_ws, size_t ws_size,
                              hipStream_t stream)
{
    const float* I = (const float*)d_in[0];
    const float* p = (const float*)d_in[1];
    const int planes = in_sizes[0] / (Hh * Ww);   // 12

    // scratch: a, b planes (2 * planes * 1M floats = 96 MB)
    float* a = (float*)d_ws;
    float* b = a + (size_t)planes * Hh * Ww;
    float* out = (float*)d_out;

    dim3 grid(Ww / STRIP, Hh / SEGH, planes);     // (16, 8, 12) = 1536 waves
    dim3 block(32);                               // one wave32 per block

    guided_boxpass<1><<<grid, block, 0, stream>>>(I, p, a, b, nullptr);
    guided_boxpass<2><<<grid, block, 0, stream>>>(a, b, out, nullptr, I);
}